// VisualTransformer_18562848653413
// MI455X (gfx1250) — compile-verified
//
#include <hip/hip_runtime.h>
#include <math.h>

typedef float v2f __attribute__((ext_vector_type(2)));
typedef float v8f __attribute__((ext_vector_type(8)));

namespace {
constexpr int kPS   = 16;
constexpr int kPR   = 14;    // 224 / 16
constexpr int kN    = 196;   // 14*14 patches
constexpr int kC    = 3;
constexpr int kW    = 224;
constexpr int kBS   = 64;    // B*S
constexpr int kF    = 768;   // C*PS*PS
constexpr int kHid  = 768;
constexpr int kOut  = 768;
constexpr int kRows = kBS * kN;                 // 12544
constexpr int kTileRows     = 16;
constexpr int kWaves        = 8;                // 256 threads, wave32
constexpr int kTilesPerWave = (kHid / 16) / kWaves; // 6
}

__device__ __forceinline__ float gelu_exact(float v) {
  return 0.5f * v * (1.0f + erff(v * 0.70710678118654752440f));
}

// ---------------------------------------------------------------------------
// Kernel 1: conv3x3 + bias + BN + exact GELU + patchify + mask*M -> flat[rows][768]
// Bandwidth-bound prologue; one thread per conv output element.
// ---------------------------------------------------------------------------
__global__ __launch_bounds__(256)
void conv_bn_gelu_patch(const float* __restrict__ x,
                        const float* __restrict__ cw,
                        const float* __restrict__ cb,
                        const float* __restrict__ bng,
                        const float* __restrict__ bnb,
                        const float* __restrict__ bnm,
                        const float* __restrict__ bnv,
                        const float* __restrict__ M,
                        const int*   __restrict__ mask,
                        float* __restrict__ flat) {
  const int total = kBS * kC * kW * kW;
  int idx = blockIdx.x * blockDim.x + threadIdx.x;
  if (idx >= total) return;

  const int h   = idx % kW;
  const int w   = (idx / kW) % kW;
  const int co  = (idx / (kW * kW)) % kC;
  const int img = idx / (kW * kW * kC);

  float acc = cb[co];
#pragma unroll
  for (int ci = 0; ci < kC; ++ci) {
    const float* __restrict__ xp = x + ((size_t)(img * kC + ci)) * kW * kW;
#pragma unroll
    for (int kw = 0; kw < 3; ++kw) {
      const int ww = w + kw - 1;
      if (ww < 0 || ww >= kW) continue;
#pragma unroll
      for (int kh = 0; kh < 3; ++kh) {
        const int hh = h + kh - 1;
        if (hh < 0 || hh >= kW) continue;
        acc += xp[ww * kW + hh] * cw[((co * kC + ci) * 3 + kw) * 3 + kh];
      }
    }
  }
  // BatchNorm (inference)
  const float inv = bng[co] * rsqrtf(bnv[co] + 1e-5f);
  acc = (acc - bnm[co]) * inv + bnb[co];
  acc = gelu_exact(acc);

  // Patchify: n = pr_w*14 + pr_h ; feature f = c*256 + pi*16 + pj
  const int pi = w & 15, pw = w >> 4;
  const int pj = h & 15, ph = h >> 4;
  const int n  = pw * kPR + ph;
  if (mask[n] > 0) acc *= M[pi * kPS + pj];

  const int row = img * kN + n;
  const int f   = co * kPS * kPS + pi * kPS + pj;
  flat[(size_t)row * kF + f] = acc;
}

// ---------------------------------------------------------------------------
// Kernel 2: T = GELU(LN(flat @ W1^T + b1))
// One block = 16x768 output panel. A-panel (16x768 f32, 48KB) staged in LDS,
// reused as the output tile for the cross-wave LayerNorm.
// fp32 WMMA: V_WMMA_F32_16X16X4_F32, K stepped by 4.
// ---------------------------------------------------------------------------
__global__ __launch_bounds__(256, 2)
void gemm1_ln_gelu(const float* __restrict__ A,    // flat [kRows][kF]
                   const float* __restrict__ W1,   // [kHid][kF]
                   const float* __restrict__ b1,
                   const float* __restrict__ lng,
                   const float* __restrict__ lnb,
                   float* __restrict__ T) {        // [kRows][kHid]
  __shared__ float sA[kTileRows * kF];
  __shared__ float sMu[kTileRows];
  __shared__ float sRs[kTileRows];

  const int tid  = threadIdx.x;
  const int wave = tid >> 5;
  const int lane = tid & 31;
  const int hf   = lane >> 4;   // half-wave select
  const int l16  = lane & 15;
  const size_t row0 = (size_t)blockIdx.x * kTileRows;

  // A rows are contiguous: copy the whole 16x768 panel coalesced.
  for (int i = tid; i < kTileRows * kF; i += 256) sA[i] = A[row0 * kF + i];
  __syncthreads();

  const int kOff = 2 * hf;  // lanes 0-15 hold K={k,k+1}, lanes 16-31 K={k+2,k+3}
  v8f acc[kTilesPerWave];

  for (int t = 0; t < kTilesPerWave; ++t) {
    v8f c = {};
    const int n = (wave * kTilesPerWave + t) * 16 + l16;
    const float* __restrict__ bcol = W1 + (size_t)n * kF + kOff;  // B[k][n] = W1[n][k]
    const float* __restrict__ arow = sA + l16 * kF + kOff;
    for (int k = 0; k < kF; k += 4) {
      v2f a = *(const v2f*)(arow + k);
      v2f b = *(const v2f*)(bcol + k);
      c = __builtin_amdgcn_wmma_f32_16x16x4_f32(false, a, false, b,
                                                (short)0, c, false, false);
    }
    acc[t] = c;
  }
  __syncthreads();  // all waves done reading sA; reuse it as the output tile

  // Scatter D (+bias) into sA[m][n]: m = reg + 8*hf, n per-lane column.
  for (int t = 0; t < kTilesPerWave; ++t) {
    const int n = (wave * kTilesPerWave + t) * 16 + l16;
    const float bias = b1[n];
#pragma unroll
    for (int r = 0; r < 8; ++r) {
      sA[(r + 8 * hf) * kF + n] = acc[t][r] + bias;
    }
  }
  __syncthreads();

  // LayerNorm stats: one thread per row (tiny vs. the GEMM).
  if (tid < kTileRows) {
    float s = 0.f, s2 = 0.f;
    const float* rp = sA + tid * kF;
    for (int c = 0; c < kF; ++c) { const float v = rp[c]; s += v; s2 += v * v; }
    const float mu  = s / kF;
    const float var = s2 / kF - mu * mu;
    sMu[tid] = mu;
    sRs[tid] = rsqrtf(var + 1e-5f);
  }
  __syncthreads();

  // Apply LN + GELU, store coalesced.
  for (int i = tid; i < kTileRows * kF; i += 256) {
    const int r = i / kF, c = i % kF;
    const float v = (sA[i] - sMu[r]) * sRs[r] * lng[c] + lnb[c];
    T[row0 * kF + i] = gelu_exact(v);
  }
}

// ---------------------------------------------------------------------------
// Kernel 3: out = T[perm-rows] @ W2^T + b2 ; permutation folded into A gather.
// ---------------------------------------------------------------------------
__global__ __launch_bounds__(256, 2)
void gemm2_perm(const float* __restrict__ T,    // [kRows][kHid]
                const float* __restrict__ W2,   // [kOut][kHid]
                const float* __restrict__ b2,
                const int*   __restrict__ perm,
                float* __restrict__ out) {      // [kRows][kOut]
  __shared__ float sA[kTileRows * kF];
  __shared__ int   sRow[kTileRows];

  const int tid  = threadIdx.x;
  const int wave = tid >> 5;
  const int lane = tid & 31;
  const int hf   = lane >> 4;
  const int l16  = lane & 15;
  const int row0 = blockIdx.x * kTileRows;

  if (tid < kTileRows) {
    const int g   = row0 + tid;
    const int img = g / kN;
    const int n   = g - img * kN;
    sRow[tid] = img * kN + perm[n];   // gather source row
  }
  __syncthreads();
  for (int i = tid; i < kTileRows * kF; i += 256) {
    const int r = i / kF, c = i % kF;
    sA[i] = T[(size_t)sRow[r] * kF + c];
  }
  __syncthreads();

  const int kOff = 2 * hf;
  for (int t = 0; t < kTilesPerWave; ++t) {
    v8f c = {};
    const int n = (wave * kTilesPerWave + t) * 16 + l16;
    const float* __restrict__ bcol = W2 + (size_t)n * kF + kOff;  // B[k][n] = W2[n][k]
    const float* __restrict__ arow = sA + l16 * kF + kOff;
    for (int k = 0; k < kF; k += 4) {
      v2f a = *(const v2f*)(arow + k);
      v2f b = *(const v2f*)(bcol + k);
      c = __builtin_amdgcn_wmma_f32_16x16x4_f32(false, a, false, b,
                                                (short)0, c, false, false);
    }
    const float bias = b2[n];
#pragma unroll
    for (int r = 0; r < 8; ++r) {
      const int m = r + 8 * hf;
      out[(size_t)(row0 + m) * kOut + n] = c[r] + bias;
    }
  }
}

// ---------------------------------------------------------------------------
extern "C" void kernel_launch(void* const* d_in, const int* in_sizes, int n_in,
                              void* d_out, int out_size, void* d_ws, size_t ws_size,
                              hipStream_t stream) {
  const float* x    = (const float*)d_in[0];
  const float* cw   = (const float*)d_in[1];
  const float* cb   = (const float*)d_in[2];
  const float* bng  = (const float*)d_in[3];
  const float* bnb  = (const float*)d_in[4];
  const float* bnm  = (const float*)d_in[5];
  const float* bnv  = (const float*)d_in[6];
  const float* M    = (const float*)d_in[7];
  const float* W1   = (const float*)d_in[8];
  const float* b1   = (const float*)d_in[9];
  const float* lng  = (const float*)d_in[10];
  const float* lnb  = (const float*)d_in[11];
  const float* W2   = (const float*)d_in[12];
  const float* b2   = (const float*)d_in[13];
  const int*   mask = (const int*)d_in[14];
  const int*   perm = (const int*)d_in[15];
  float* out = (float*)d_out;

  float* flat = (float*)d_ws;                       // 12544*768 f32 (~38.5 MB)
  float* Tbuf = flat + (size_t)kRows * kF;          // 12544*768 f32 (~38.5 MB)

  {
    const int total = kBS * kC * kW * kW;           // 9,633,792
    conv_bn_gelu_patch<<<(total + 255) / 256, 256, 0, stream>>>(
        x, cw, cb, bng, bnb, bnm, bnv, M, mask, flat);
  }
  gemm1_ln_gelu<<<kRows / kTileRows, 256, 0, stream>>>(flat, W1, b1, lng, lnb, Tbuf);
  gemm2_perm  <<<kRows / kTileRows, 256, 0, stream>>>(Tbuf, W2, b2, perm, out);
}